// Linear_VQ_VAE_49082886259062
// MI455X (gfx1250) — compile-verified
//
#include <hip/hip_runtime.h>
#include <hip/hip_bf16.h>
#include <math.h>

typedef float v2f __attribute__((ext_vector_type(2)));
typedef float v8f __attribute__((ext_vector_type(8)));
typedef int   v4i __attribute__((ext_vector_type(4)));

#define NF     4096     // feature dim (square layers)
#define NB     4096     // batch
#define NEMB   512
#define EDIM   64
#define NCODES ((NB * NF) / EDIM)   // 262144

// ---------------- GEMM: Y = X @ W^T + b (optional ReLU) ----------------
// X:[M,K] row-major, W:[N,K] row-major (so B[k][n] = W[n][k]), Y:[M,N]
#define TM  128
#define TN  128
#define TK  32
#define TKP 36          // padded K-stride (words): 16 lanes * stride 36 tile all 64 banks

#if defined(__gfx1250__) && __has_builtin(__builtin_amdgcn_global_load_async_to_lds_b128) && __has_builtin(__builtin_amdgcn_s_wait_asynccnt)
#define USE_ASYNC 1
// builtin params: (AS1 int4* gsrc, AS3 int4* ldst, imm offset, imm cpol)
#define GPTR(p) ((__attribute__((address_space(1))) v4i*)(p))
#define LPTR(p) ((__attribute__((address_space(3))) v4i*)(p))
#else
#define USE_ASYNC 0
#endif

// Stage one 128x32 tile of X (rows=M) and one of W (rows=N) into LDS,
// verbatim row-major (K contiguous) so async 16B copies need no shuffling.
__device__ __forceinline__ void stage_slab(
    const float* __restrict__ X, const float* __restrict__ W,
    float (*As)[TKP], float (*Bs)[TKP],
    int bm, int bn, int k0, int tid)
{
  #pragma unroll
  for (int it = 0; it < 4; ++it) {           // 1024 float4 per tile (A and B each) / 256 thr
    int idx = tid + it * 256;
    int r = idx >> 3;                         // 0..127
    int c = (idx & 7) * 4;                    // 0,4,...,28
    const float* ga = X + (size_t)(bm + r) * NF + k0 + c;
    const float* gb = W + (size_t)(bn + r) * NF + k0 + c;
#if USE_ASYNC
    __builtin_amdgcn_global_load_async_to_lds_b128(GPTR(ga), LPTR(&As[r][c]), 0, 0);
    __builtin_amdgcn_global_load_async_to_lds_b128(GPTR(gb), LPTR(&Bs[r][c]), 0, 0);
#else
    *(float4*)&As[r][c] = *(const float4*)ga;
    *(float4*)&Bs[r][c] = *(const float4*)gb;
#endif
  }
}

__global__ __launch_bounds__(256) void gemm_bias_kernel(
    const float* __restrict__ X, const float* __restrict__ W,
    const float* __restrict__ bias, float* __restrict__ Y, int relu)
{
  __shared__ float As[2][TM][TKP];   // 2 x 18KB
  __shared__ float Bs[2][TN][TKP];   // 2 x 18KB  (row = N index, K contiguous)

  const int tid  = threadIdx.x;
  const int wave = tid >> 5;
  const int lane = tid & 31;
  const int half = lane >> 4;        // 0: lanes 0-15, 1: lanes 16-31
  const int l15  = lane & 15;
  const int bm   = blockIdx.y * TM;
  const int bn   = blockIdx.x * TN;
  const int wm   = (wave >> 1) * 32; // wave tile: 32(M) x 64(N), 4x2 wave grid
  const int wn   = (wave & 1) * 64;

  v8f acc[2][4];
  #pragma unroll
  for (int mt = 0; mt < 2; ++mt)
    #pragma unroll
    for (int t = 0; t < 4; ++t) { v8f z = {}; acc[mt][t] = z; }

  // Prologue: stage slab 0
  stage_slab(X, W, As[0], Bs[0], bm, bn, 0, tid);
#if USE_ASYNC
  __builtin_amdgcn_s_wait_asynccnt(0);
#endif
  __syncthreads();

  for (int k0 = 0; k0 < NF; k0 += TK) {
    const int buf = (k0 / TK) & 1;
    if (k0 + TK < NF)
      stage_slab(X, W, As[buf ^ 1], Bs[buf ^ 1], bm, bn, k0 + TK, tid);

    #pragma unroll
    for (int kk = 0; kk < TK / 4; ++kk) {
      // WMMA f32 16x16x4 fragments: VGPR0=K{0|2}, VGPR1=K{1|3} per half-wave.
      // {K,K+1} adjacent in LDS -> single aligned ds_load_b64 per operand.
      const int kb = kk * 4 + half * 2;
      v2f a0 = *(const v2f*)&As[buf][wm + l15][kb];
      v2f a1 = *(const v2f*)&As[buf][wm + 16 + l15][kb];
      #pragma unroll
      for (int t = 0; t < 4; ++t) {
        v2f bf = *(const v2f*)&Bs[buf][wn + t * 16 + l15][kb];
        acc[0][t] = __builtin_amdgcn_wmma_f32_16x16x4_f32(
            false, a0, false, bf, (short)0, acc[0][t], false, false);
        acc[1][t] = __builtin_amdgcn_wmma_f32_16x16x4_f32(
            false, a1, false, bf, (short)0, acc[1][t], false, false);
      }
    }

#if USE_ASYNC
    __builtin_amdgcn_s_wait_asynccnt(0);   // next slab fully in LDS
#endif
    __syncthreads();                       // all waves done with 'buf'
  }

  // Epilogue: C/D layout -> VGPR r holds rows (r | r+8) per half-wave
  #pragma unroll
  for (int mt = 0; mt < 2; ++mt) {
    #pragma unroll
    for (int t = 0; t < 4; ++t) {
      const int col = bn + wn + t * 16 + l15;
      const float bv = bias[col];
      #pragma unroll
      for (int r = 0; r < 8; ++r) {
        const int row = bm + wm + mt * 16 + half * 8 + r;
        float v = acc[mt][t][r] + bv;
        if (relu) v = fmaxf(v, 0.0f);
        Y[(size_t)row * NF + col] = v;
      }
    }
  }
}

// ---------------- Vector quantizer ----------------
#define VQ_CHUNK 128   // codebook entries staged in LDS per pass (32KB)

__global__ __launch_bounds__(256) void vq_kernel(
    const float* __restrict__ latent, const float* __restrict__ codebook,
    float* __restrict__ quant, float* __restrict__ g_counts,
    float* __restrict__ g_sqerr)
{
  __shared__ float cb[VQ_CHUNK * EDIM];
  __shared__ float ee[VQ_CHUNK];
  __shared__ float hist[NEMB];
  __shared__ float blk_sqerr;

  const int tid = threadIdx.x;
  for (int k = tid; k < NEMB; k += 256) hist[k] = 0.0f;
  if (tid == 0) blk_sqerr = 0.0f;

  const int row = blockIdx.x * 256 + tid;
  float4 cv[16];
  const float4* cp = (const float4*)(latent + (size_t)row * EDIM);
  float cc = 0.0f;
  #pragma unroll
  for (int j = 0; j < 16; ++j) {
    cv[j] = cp[j];
    cc += cv[j].x * cv[j].x + cv[j].y * cv[j].y +
          cv[j].z * cv[j].z + cv[j].w * cv[j].w;
  }

  float bestd = 3.4e38f;
  int   bestk = 0;

  for (int c0 = 0; c0 < NEMB; c0 += VQ_CHUNK) {
    __syncthreads();
    for (int i = tid; i < VQ_CHUNK * EDIM / 4; i += 256)
      ((float4*)cb)[i] = ((const float4*)(codebook + (size_t)c0 * EDIM))[i];
    __syncthreads();
    for (int k = tid; k < VQ_CHUNK; k += 256) {
      float s = 0.0f;
      #pragma unroll 8
      for (int j = 0; j < EDIM; ++j) { float e = cb[k * EDIM + j]; s += e * e; }
      ee[k] = s;
    }
    __syncthreads();

    for (int k = 0; k < VQ_CHUNK; ++k) {
      const float4* ep = (const float4*)(cb + k * EDIM);
      float dot = 0.0f;
      #pragma unroll
      for (int j = 0; j < 16; ++j) {
        float4 e = ep[j];
        dot += cv[j].x * e.x + cv[j].y * e.y + cv[j].z * e.z + cv[j].w * e.w;
      }
      float d = ee[k] - 2.0f * dot;       // same argmin as ||c-e||^2
      if (d < bestd) { bestd = d; bestk = c0 + k; }
    }
  }

  // quantized row (straight-through forward value == codebook[bestk])
  float4* qp = (float4*)(quant + (size_t)row * EDIM);
  const float4* bp = (const float4*)(codebook + (size_t)bestk * EDIM);
  #pragma unroll
  for (int j = 0; j < 16; ++j) qp[j] = bp[j];

  atomicAdd(&hist[bestk], 1.0f);
  atomicAdd(&blk_sqerr, cc + bestd);      // = ||c - e_best||^2
  __syncthreads();
  if (tid == 0) atomicAdd(g_sqerr, blk_sqerr);
  for (int k = tid; k < NEMB; k += 256) {
    float h = hist[k];
    if (h != 0.0f) atomicAdd(&g_counts[k], h);
  }
}

// ---------------- Scalars ----------------
__global__ void zero_kernel(float* __restrict__ p, int n) {
  int i = blockIdx.x * blockDim.x + threadIdx.x;
  if (i < n) p[i] = 0.0f;
}

__global__ __launch_bounds__(512) void finalize_kernel(
    const float* __restrict__ g_counts, const float* __restrict__ g_sqerr,
    float* __restrict__ out_loss, float* __restrict__ out_perp)
{
  __shared__ float red[512];
  const int tid = threadIdx.x;
  float p = g_counts[tid] / (float)NCODES;
  red[tid] = p * logf(p + 1e-10f);
  __syncthreads();
  for (int s = 256; s > 0; s >>= 1) {
    if (tid < s) red[tid] += red[tid + s];
    __syncthreads();
  }
  if (tid == 0) {
    *out_perp = expf(-red[0]);
    // forward value: q_latent_loss == e_latent_loss -> (1 + 0.25) * mse
    *out_loss = 1.25f * (g_sqerr[0] / (float)((size_t)NCODES * EDIM));
  }
}

// ---------------- Launcher ----------------
extern "C" void kernel_launch(void* const* d_in, const int* in_sizes, int n_in,
                              void* d_out, int out_size, void* d_ws, size_t ws_size,
                              hipStream_t stream) {
  (void)in_sizes; (void)n_in; (void)out_size; (void)ws_size;
  const float* x  = (const float*)d_in[0];
  const float* W1 = (const float*)d_in[1];
  const float* b1 = (const float*)d_in[2];
  const float* W2 = (const float*)d_in[3];
  const float* b2 = (const float*)d_in[4];
  const float* W3 = (const float*)d_in[5];
  const float* b3 = (const float*)d_in[6];
  const float* W4 = (const float*)d_in[7];
  const float* b4 = (const float*)d_in[8];
  const float* cb = (const float*)d_in[9];
  float* out = (float*)d_out;

  const size_t big = (size_t)NB * NF;          // 16M floats = 64MB
  float* buf0   = (float*)d_ws;                // activations ping
  float* buf1   = buf0 + big;                  // activations pong
  float* counts = buf1 + big;                  // 512 floats
  float* sqerr  = counts + NEMB;               // 1 float

  zero_kernel<<<(NEMB + 1 + 255) / 256, 256, 0, stream>>>(counts, NEMB + 1);

  dim3 grid(NF / TN, NB / TM);                 // 32 x 32 tiles
  gemm_bias_kernel<<<grid, 256, 0, stream>>>(x,    W1, b1, buf0, 1); // fc1+relu
  gemm_bias_kernel<<<grid, 256, 0, stream>>>(buf0, W2, b2, buf1, 0); // fc2
  vq_kernel<<<NCODES / 256, 256, 0, stream>>>(buf1, cb, buf0, counts, sqerr);
  gemm_bias_kernel<<<grid, 256, 0, stream>>>(buf0, W3, b3, buf1, 1); // fc3+relu
  gemm_bias_kernel<<<grid, 256, 0, stream>>>(buf1, W4, b4, out,  0); // fc4
  finalize_kernel<<<1, 512, 0, stream>>>(counts, sqerr, out + big, out + big + 1);
}